// AttentionBlock_65481071396323
// MI455X (gfx1250) — compile-verified
//
#include <hip/hip_runtime.h>
#include <math.h>

typedef __attribute__((ext_vector_type(2))) float v2f;
typedef __attribute__((ext_vector_type(8))) float v8f;

#define CDIM 256      // channels
#define CRDIM 64      // reduced channels (C/R)
#define BATCH 32
#define HW    12544   // 112*112
#define HW4   3136    // HW / 4

// ---------------------------------------------------------------------------
// Kernel 1: global average pool. One 256-thread block per (b, c) plane.
// Streams the 50 KB plane with float4 loads, LDS tree reduction.
// ---------------------------------------------------------------------------
__global__ __launch_bounds__(256)
void gap_kernel(const float* __restrict__ x, float* __restrict__ gap) {
    __shared__ float red[256];
    const int plane = blockIdx.x;                      // b*256 + c
    const float4* p = (const float4*)(x + (size_t)plane * HW);
    float s = 0.f;
    for (int i = threadIdx.x; i < HW4; i += 256) {
        float4 v = p[i];
        s += (v.x + v.y) + (v.z + v.w);
    }
    red[threadIdx.x] = s;
    __syncthreads();
    for (int off = 128; off > 0; off >>= 1) {
        if (threadIdx.x < off) red[threadIdx.x] += red[threadIdx.x + off];
        __syncthreads();
    }
    if (threadIdx.x == 0) gap[plane] = red[0] * (1.0f / (float)HW);
}

// ---------------------------------------------------------------------------
// Kernel 2: the SE bottleneck, single block, 8 wave32s, fp32 WMMA.
//   h    = relu(gap[32x256] x w1^T[256x64] + b1)      (FC1)
//   gate = sigmoid(h[32x64] x w2^T[64x256] + b2)      (FC2)
// V_WMMA_F32_16X16X4_F32 fragment layouts (ISA 7.12.2):
//   A 16x4 : lane<16 -> (K=k0,k0+1), lane>=16 -> (K=k0+2,k0+3), M = lane%16
//   B 4x16 : same K split, N = lane%16
//   C/D    : VGPR r -> M = r + (lane>=16 ? 8 : 0), N = lane%16
// ---------------------------------------------------------------------------
__global__ __launch_bounds__(256, 1)
void se_fc_kernel(const float* __restrict__ gap,
                  const float* __restrict__ w1, const float* __restrict__ b1,
                  const float* __restrict__ w2, const float* __restrict__ b2,
                  float* __restrict__ gate) {
    __shared__ float hbuf[BATCH * CRDIM];              // 32x64 = 8 KB

    const int tid   = threadIdx.x;
    const int wave  = tid >> 5;
    const int lane  = tid & 31;
    const int lh    = lane & 15;                       // 0..15
    const int khi   = (lane >> 4) << 1;                // 0 or 2
    const int mskew = (lane >> 4) << 3;                // 0 or 8 (C/D row offset)

    // ---- FC1: one 16x16 tile per wave (2 M-tiles x 4 N-tiles) -------------
    {
        const int mt = wave >> 2;                      // 0..1
        const int nt = wave & 3;                       // 0..3
        const int m  = mt * 16 + lh;
        const int n  = nt * 16 + lh;
        v8f acc = {};
        for (int k0 = 0; k0 < CDIM; k0 += 4) {
            const v2f a = *(const v2f*)(gap + m * CDIM + k0 + khi);
            const v2f b = *(const v2f*)(w1  + n * CDIM + k0 + khi);
            acc = __builtin_amdgcn_wmma_f32_16x16x4_f32(
                false, a, false, b, (short)0, acc, false, false);
        }
        const float bias = b1[n];
#pragma unroll
        for (int r = 0; r < 8; ++r) {
            float v = acc[r] + bias;
            v = v > 0.f ? v : 0.f;                     // ReLU
            const int mrow = mt * 16 + r + mskew;
            hbuf[mrow * CRDIM + n] = v;
        }
    }
    __syncthreads();

    // ---- FC2: 32 tiles (2 M-tiles x 16 N-tiles), 4 tiles per wave ---------
    for (int t = wave * 4; t < wave * 4 + 4; ++t) {
        const int mt = t >> 4;                         // 0..1
        const int nt = t & 15;                         // 0..15
        const int m  = mt * 16 + lh;
        const int n  = nt * 16 + lh;
        v8f acc = {};
        for (int k0 = 0; k0 < CRDIM; k0 += 4) {
            const v2f a = *(const v2f*)(hbuf + m * CRDIM + k0 + khi);
            const v2f b = *(const v2f*)(w2   + n * CRDIM + k0 + khi);
            acc = __builtin_amdgcn_wmma_f32_16x16x4_f32(
                false, a, false, b, (short)0, acc, false, false);
        }
        const float bias = b2[n];
#pragma unroll
        for (int r = 0; r < 8; ++r) {
            float v = acc[r] + bias;
            v = 1.0f / (1.0f + __expf(-v));            // sigmoid
            const int mrow = mt * 16 + r + mskew;
            gate[mrow * CDIM + n] = v;
        }
    }
}

// ---------------------------------------------------------------------------
// Kernel 3: out = x * gate[b,c]. One block per plane, float4 streaming.
// ---------------------------------------------------------------------------
__global__ __launch_bounds__(256)
void scale_kernel(const float* __restrict__ x, const float* __restrict__ gate,
                  float* __restrict__ out) {
    const int plane = blockIdx.x;                      // b*256 + c
    const float g = gate[plane];
    const float4* xin = (const float4*)(x + (size_t)plane * HW);
    float4* o = (float4*)(out + (size_t)plane * HW);
    for (int i = threadIdx.x; i < HW4; i += 256) {
        float4 v = xin[i];
        v.x *= g; v.y *= g; v.z *= g; v.w *= g;
        o[i] = v;
    }
}

// ---------------------------------------------------------------------------
extern "C" void kernel_launch(void* const* d_in, const int* in_sizes, int n_in,
                              void* d_out, int out_size, void* d_ws, size_t ws_size,
                              hipStream_t stream) {
    const float* x  = (const float*)d_in[0];
    const float* w1 = (const float*)d_in[1];
    const float* b1 = (const float*)d_in[2];
    const float* w2 = (const float*)d_in[3];
    const float* b2 = (const float*)d_in[4];
    float* out = (float*)d_out;

    float* gap  = (float*)d_ws;                        // 32*256 floats (32 KB)
    float* gate = gap + BATCH * CDIM;                  // 32*256 floats (32 KB)

    gap_kernel  <<<BATCH * CDIM, 256, 0, stream>>>(x, gap);
    se_fc_kernel<<<1,            256, 0, stream>>>(gap, w1, b1, w2, b2, gate);
    scale_kernel<<<BATCH * CDIM, 256, 0, stream>>>(x, gate, out);
}